// funcKAN_7722351198984
// MI455X (gfx1250) — compile-verified
//
#include <hip/hip_runtime.h>
#include <hip/hip_bf16.h>

#define IN_FEATURES  2048
#define OUT_FEATURES 2048
#define ORDER        16
#define ISPLIT       16                      // i-dimension splits (blocks along y)
#define OTILES       (OUT_FEATURES / 16)     // 128 o-tiles of 16 outputs
#define WAVES_PER_BLOCK 8
#define WS_ROWS      (ISPLIT * WAVES_PER_BLOCK)   // 128 partial rows

typedef __attribute__((ext_vector_type(2))) float v2f;
typedef __attribute__((ext_vector_type(4))) float v4f;
typedef __attribute__((ext_vector_type(8))) float v8f;

// float4-granularity stride between consecutive i's in weight[i][o][k]
#define ISTRIDE4 ((size_t)OUT_FEATURES * ORDER / 4)   // 8192 float4 = 128 KB

// Sum weight[i, o, 0..15]: 4x non-temporal b128 loads (single-use stream,
// keep it out of L2's way) + 15 adds. `p` is precomputed so the compile-time
// constant element offset folds into the 24-bit instruction offset.
__device__ __forceinline__ float ksum16(const v4f* __restrict__ p) {
    v4f a = __builtin_nontemporal_load(p + 0);
    v4f b = __builtin_nontemporal_load(p + 1);
    v4f c = __builtin_nontemporal_load(p + 2);
    v4f d = __builtin_nontemporal_load(p + 3);
    return (((a[0] + a[1]) + (a[2] + a[3])) + ((b[0] + b[1]) + (b[2] + b[3])))
         + (((c[0] + c[1]) + (c[2] + c[3])) + ((d[0] + d[1]) + (d[2] + d[3])));
}

// Phase 1: each wave contracts its 16 i's against a 16-wide o-tile using
// V_WMMA_F32_16X16X4_F32 (kk = 4 i's per instruction, C accumulates),
// writing 16 partial outputs to a private workspace row.
__global__ void __launch_bounds__(256)
funcKAN_partial_kernel(const float* __restrict__ x,
                       const float* __restrict__ weight,
                       float* __restrict__ ws) {
    const int lane   = threadIdx.x & 31;
    const int wave   = threadIdx.x >> 5;          // 0..7
    const int o_tile = blockIdx.x;                // 0..127
    const int isplit = blockIdx.y;                // 0..15
    const int o_base = o_tile * 16;
    const int i_base = isplit * (IN_FEATURES / ISPLIT) + wave * 16;  // 16 i's per wave

    const int  lo    = lane & 15;
    const int  half  = lane >> 4;                 // 0 for lanes 0-15, 1 for lanes 16-31
    const int  o     = o_base + lo;

    // One 64-bit base per lane; every load in the unrolled loop is
    // base + constant (max ~1.8 MB, fits the signed 24-bit IOFFSET).
    const v4f* base = (const v4f*)(weight
        + ((size_t)(i_base + half) * OUT_FEATURES + (size_t)o) * ORDER);

    v8f c = {};                                    // D rows all identical (A rows equal)

    #pragma unroll
    for (int r = 0; r < 4; ++r) {
        const int i0 = i_base + r * 4;

        // Sub-round A: lanes 0-15 -> i0,   lanes 16-31 -> i0+1
        float sA = ksum16(base + (size_t)(4 * r)     * ISTRIDE4);
        // Sub-round B: lanes 0-15 -> i0+2, lanes 16-31 -> i0+3
        float sB = ksum16(base + (size_t)(4 * r + 2) * ISTRIDE4);

        // B-matrix layout (4x16 f32, 2 VGPRs):
        //   v0: lanes 0-15 = row kk=0 (s[i0,n]),  lanes 16-31 = row kk=2 (s[i2,n])
        //   v1: lanes 0-15 = row kk=1 (s[i1,n]),  lanes 16-31 = row kk=3 (s[i3,n])
        int pB = __builtin_amdgcn_ds_bpermute(lo * 4,        __float_as_int(sB)); // s[i2, n]
        int pA = __builtin_amdgcn_ds_bpermute((lo + 16) * 4, __float_as_int(sA)); // s[i1, n]
        v2f B;
        B[0] = (lane < 16) ? sA : __int_as_float(pB);
        B[1] = (lane < 16) ? __int_as_float(pA) : sB;

        // A-matrix (16x4 f32): A[m][kk] = x[i_kk]/ORDER, broadcast over all m rows.
        //   v0: lanes 0-15 = kk=0, lanes 16-31 = kk=2 ; v1: kk=1 / kk=3
        float xs0 = x[i0 + 0] * (1.0f / ORDER);
        float xs1 = x[i0 + 1] * (1.0f / ORDER);
        float xs2 = x[i0 + 2] * (1.0f / ORDER);
        float xs3 = x[i0 + 3] * (1.0f / ORDER);
        v2f A;
        A[0] = (lane < 16) ? xs0 : xs2;
        A[1] = (lane < 16) ? xs1 : xs3;

        // D = A x B + C  (16x16x4 f32 WMMA), accumulate over the 4 rounds.
        c = __builtin_amdgcn_wmma_f32_16x16x4_f32(
                /*neg_a=*/false, A, /*neg_b=*/false, B,
                /*c_mod=*/(short)0, c, /*reuse_a=*/false, /*reuse_b=*/false);
    }

    // C/D layout: VGPR0, lanes 0-15 hold D[M=0][N=lane] = partial out[o_base+lane].
    if (lane < 16) {
        const int row = isplit * WAVES_PER_BLOCK + wave;   // unique row per wave
        ws[(size_t)row * OUT_FEATURES + o_base + lane] = c[0];
    }
}

// Phase 2: deterministic fixed-order reduction of the 128 partial rows.
__global__ void __launch_bounds__(256)
funcKAN_reduce_kernel(const float* __restrict__ ws, float* __restrict__ out) {
    const int o = blockIdx.x * blockDim.x + threadIdx.x;
    if (o >= OUT_FEATURES) return;
    float acc = 0.0f;
    #pragma unroll 8
    for (int r = 0; r < WS_ROWS; ++r) {
        acc += ws[(size_t)r * OUT_FEATURES + o];
    }
    out[o] = acc;
}

extern "C" void kernel_launch(void* const* d_in, const int* in_sizes, int n_in,
                              void* d_out, int out_size, void* d_ws, size_t ws_size,
                              hipStream_t stream) {
    const float* x      = (const float*)d_in[0];   // [2048]
    const float* weight = (const float*)d_in[1];   // [2048, 2048, 16]
    float*       out    = (float*)d_out;           // [2048]
    float*       ws     = (float*)d_ws;            // needs 128*2048*4 = 1 MB

    dim3 grid1(OTILES, ISPLIT);                    // 128 x 16 = 2048 blocks
    funcKAN_partial_kernel<<<grid1, 256, 0, stream>>>(x, weight, ws);

    dim3 grid2((OUT_FEATURES + 255) / 256);        // 8 blocks
    funcKAN_reduce_kernel<<<grid2, 256, 0, stream>>>(ws, out);
}